// AdaptiveFocusedLoss_10857677325105
// MI455X (gfx1250) — compile-verified
//
#include <hip/hip_runtime.h>
#include <stdint.h>

#define B_TOTAL 524288
#define NCLS    128
#define TPB     256
#define WPB     8                       // waves per block (wave32)
#define NBLK    2048
#define NWAVES  (NBLK * WPB)            // 16384 waves
#define ITERS   (B_TOTAL / NWAVES)      // 32 rows per wave

__device__ __forceinline__ float wave_reduce_max(float v) {
#pragma unroll
    for (int off = 16; off > 0; off >>= 1) v = fmaxf(v, __shfl_xor(v, off, 32));
    return v;
}
__device__ __forceinline__ float wave_reduce_sum(float v) {
#pragma unroll
    for (int off = 16; off > 0; off >>= 1) v += __shfl_xor(v, off, 32);
    return v;
}
__device__ __forceinline__ unsigned wave_reduce_sum_u(unsigned v) {
#pragma unroll
    for (int off = 16; off > 0; off >>= 1) v += (unsigned)__shfl_xor((int)v, off, 32);
    return v;
}

__global__ void afl_init(double* acc, unsigned long long* cnt) {
    acc[0] = 0.0;   // base-loss sum
    acc[1] = 0.0;   // penalty sum
    *cnt   = 0ull;  // mask count
}

__global__ __launch_bounds__(TPB) void afl_main(
    const float* __restrict__ outputs,
    const int*   __restrict__ labels,
    const float* __restrict__ wts,
    double* __restrict__ acc,
    unsigned long long* __restrict__ cnt)
{
    __shared__ __align__(16) float wsh[NCLS * NCLS];   // 64 KB confusion table
    __shared__ float    s_base[WPB], s_pen[WPB];
    __shared__ unsigned s_cnt[WPB];

    // ---- CDNA5 async global->LDS broadcast of the 64KB table ----
    // 256 threads x b128 x 16 rounds = 65536 B. Tracked on ASYNCcnt.
    {
        unsigned lds_base = (unsigned)(uintptr_t)(void*)wsh;   // flat addr[31:0] == LDS offset
        unsigned ldso = lds_base + threadIdx.x * 16u;
        unsigned long long gp = (unsigned long long)(uintptr_t)wts
                              + (unsigned long long)(threadIdx.x * 16u);
#pragma unroll
        for (int k = 0; k < 16; ++k) {
            asm volatile("global_load_async_to_lds_b128 %0, %1, off"
                         :: "v"(ldso + (unsigned)k * 4096u),
                            "v"(gp + (unsigned long long)k * 4096ull)
                         : "memory");
        }
        asm volatile("s_wait_asynccnt 0" ::: "memory");
    }
    __syncthreads();

    const int lane  = threadIdx.x & 31;
    const int wave  = threadIdx.x >> 5;
    const int gwave = blockIdx.x * WPB + wave;

    float    base_acc = 0.0f;
    float    pen_acc  = 0.0f;
    unsigned cnt_acc  = 0u;

    for (int i = 0; i < ITERS; ++i) {
        const int r = i * NWAVES + gwave;              // contiguous 8MB sweep per iter

        int lab = labels[r];                           // wave-uniform
        lab = __builtin_amdgcn_readfirstlane(lab);

        const float4 x = *(const float4*)(outputs + (size_t)r * NCLS + lane * 4);

        if (i + 1 < ITERS)                             // global_prefetch next row
            __builtin_prefetch(outputs + (size_t)(r + NWAVES) * NCLS + lane * 4, 0, 0);

        // stable log-softmax pieces
        float m = fmaxf(fmaxf(x.x, x.y), fmaxf(x.z, x.w));
        m = wave_reduce_max(m);

        const float e0 = __expf(x.x - m), e1 = __expf(x.y - m);
        const float e2 = __expf(x.z - m), e3 = __expf(x.w - m);
        const float s  = wave_reduce_sum(e0 + e1 + e2 + e3);
        const float invs = 1.0f / s;
        const float lse  = __logf(s);

        // x[label]: owning lane = lab/4, element = lab%4
        const float xr[4] = {x.x, x.y, x.z, x.w};
        const float xl = __shfl(xr[lab & 3], lab >> 2, 32);
        if (lane == 0) base_acc += (m + lse - xl);     // -logp[label]

        // penalty terms from LDS weight row
        const float4 wv = *(const float4*)(&wsh[lab * NCLS + lane * 4]);
        const float p0 = e0 * invs, p1 = e1 * invs, p2 = e2 * invs, p3 = e3 * invs;
        const int   c0 = lane * 4;

        { const bool md = (c0 + 0 != lab) & (wv.x > 1.0f) & (p0 > 0.2f);
          pen_acc += md ? wv.x * p0 : 0.0f; cnt_acc += md; }
        { const bool md = (c0 + 1 != lab) & (wv.y > 1.0f) & (p1 > 0.2f);
          pen_acc += md ? wv.y * p1 : 0.0f; cnt_acc += md; }
        { const bool md = (c0 + 2 != lab) & (wv.z > 1.0f) & (p2 > 0.2f);
          pen_acc += md ? wv.z * p2 : 0.0f; cnt_acc += md; }
        { const bool md = (c0 + 3 != lab) & (wv.w > 1.0f) & (p3 > 0.2f);
          pen_acc += md ? wv.w * p3 : 0.0f; cnt_acc += md; }
    }

    // wave -> block -> global reduction
    base_acc = wave_reduce_sum(base_acc);
    pen_acc  = wave_reduce_sum(pen_acc);
    cnt_acc  = wave_reduce_sum_u(cnt_acc);
    if (lane == 0) { s_base[wave] = base_acc; s_pen[wave] = pen_acc; s_cnt[wave] = cnt_acc; }
    __syncthreads();

    if (wave == 0) {
        float    b = (lane < WPB) ? s_base[lane] : 0.0f;
        float    p = (lane < WPB) ? s_pen[lane]  : 0.0f;
        unsigned c = (lane < WPB) ? s_cnt[lane]  : 0u;
        b = wave_reduce_sum(b);
        p = wave_reduce_sum(p);
        c = wave_reduce_sum_u(c);
        if (lane == 0) {
            atomicAdd(&acc[0], (double)b);
            atomicAdd(&acc[1], (double)p);
            atomicAdd(cnt, (unsigned long long)c);
        }
    }
}

__global__ void afl_final(const double* __restrict__ acc,
                          const unsigned long long* __restrict__ cnt,
                          float* __restrict__ out)
{
    const double base = acc[0] / (double)B_TOTAL;
    const double pen  = (*cnt > 0) ? (acc[1] / (double)(*cnt)) : 0.0;
    out[0] = (float)(base + 0.5 * pen);
}

extern "C" void kernel_launch(void* const* d_in, const int* in_sizes, int n_in,
                              void* d_out, int out_size, void* d_ws, size_t ws_size,
                              hipStream_t stream)
{
    const float* outputs = (const float*)d_in[0];   // [B, C] f32
    const int*   labels  = (const int*)d_in[1];     // [B] i32
    const float* wts     = (const float*)d_in[2];   // [C, C] f32

    double* acc = (double*)d_ws;                                     // 2 x f64
    unsigned long long* cnt = (unsigned long long*)((char*)d_ws + 16);

    afl_init <<<1, 1, 0, stream>>>(acc, cnt);
    afl_main <<<NBLK, TPB, 0, stream>>>(outputs, labels, wts, acc, cnt);
    afl_final<<<1, 1, 0, stream>>>(acc, cnt, (float*)d_out);
}